// TinyTransformer_23227183137148
// MI455X (gfx1250) — compile-verified
//
#include <hip/hip_runtime.h>
#include <math.h>

#define DIM 12
#define FF  24
#define SEQ 8
#define VOC 128
#define TPB 256
#define ROWS_PER_BLOCK 32   // batch rows per block (32 rows * 8 tokens = 256 tokens)

typedef __attribute__((ext_vector_type(2))) float v2f;
typedef __attribute__((ext_vector_type(8))) float v8f;

__global__ __launch_bounds__(TPB) void tiny_tf_kernel(
    const int*   __restrict__ x,
    const float* __restrict__ embed,
    const float* __restrict__ pos,
    const float* __restrict__ Wq, const float* __restrict__ bq,
    const float* __restrict__ Wk, const float* __restrict__ bk,
    const float* __restrict__ Wv, const float* __restrict__ bv,
    const float* __restrict__ Wo, const float* __restrict__ bo,
    const float* __restrict__ W1, const float* __restrict__ b1,
    const float* __restrict__ W2, const float* __restrict__ b2,
    const float* __restrict__ blm,
    float* __restrict__ out)
{
    __shared__ float s_embed[VOC * DIM];
    __shared__ float s_pos[SEQ * DIM];
    __shared__ float s_Wq[DIM * DIM], s_Wk[DIM * DIM], s_Wv[DIM * DIM], s_Wo[DIM * DIM];
    __shared__ float s_bq[DIM], s_bk[DIM], s_bv[DIM], s_bo[DIM];
    __shared__ float s_W1[FF * DIM], s_b1[FF], s_W2[DIM * FF], s_b2[DIM];
    __shared__ float s_blm[VOC];
    __shared__ float s_K[TPB * DIM];
    __shared__ float s_V[TPB * DIM];
    __shared__ float s_X[TPB * DIM];   // final per-token hidden state (A-matrix for WMMA)

    const int tid = threadIdx.x;

    // ---- Stage weights into LDS ----
    for (int i = tid; i < VOC * DIM; i += TPB) s_embed[i] = embed[i];
    for (int i = tid; i < SEQ * DIM; i += TPB) s_pos[i] = pos[i];
    for (int i = tid; i < DIM * DIM; i += TPB) {
        s_Wq[i] = Wq[i]; s_Wk[i] = Wk[i]; s_Wv[i] = Wv[i]; s_Wo[i] = Wo[i];
    }
    for (int i = tid; i < FF * DIM; i += TPB) { s_W1[i] = W1[i]; s_W2[i] = W2[i]; }
    if (tid < DIM) {
        s_bq[tid] = bq[tid]; s_bk[tid] = bk[tid]; s_bv[tid] = bv[tid];
        s_bo[tid] = bo[tid]; s_b2[tid] = b2[tid];
    }
    if (tid < FF)  s_b1[tid]  = b1[tid];
    if (tid < VOC) s_blm[tid] = blm[tid];
    __syncthreads();

    // ---- Phase 1: per-token transformer body (VALU) ----
    const int n    = tid & (SEQ - 1);      // position in sequence
    const int row  = tid >> 3;             // local batch row 0..31
    const int bRow = blockIdx.x * ROWS_PER_BLOCK + row;
    const int tokBase = blockIdx.x * TPB;  // flat token index of local token 0

    float Xr[DIM], Qr[DIM];
    {
        const int tokId = x[bRow * SEQ + n];
        #pragma unroll
        for (int d = 0; d < DIM; ++d) Xr[d] = s_embed[tokId * DIM + d] + s_pos[n * DIM + d];
        #pragma unroll
        for (int d = 0; d < DIM; ++d) {
            float q = s_bq[d], k = s_bk[d], v = s_bv[d];
            #pragma unroll
            for (int c = 0; c < DIM; ++c) {
                const float xv = Xr[c];
                q += xv * s_Wq[d * DIM + c];
                k += xv * s_Wk[d * DIM + c];
                v += xv * s_Wv[d * DIM + c];
            }
            Qr[d] = q;
            s_K[tid * DIM + d] = k;
            s_V[tid * DIM + d] = v;
        }
    }
    __syncthreads();

    // causal attention over this thread's row (keys m = 0..n)
    {
        const float rscale = 0.28867513459481287f;  // 1/sqrt(12)
        const int kvBase = (tid - n) * DIM;
        float sc[SEQ];
        float mx = -INFINITY;
        for (int m = 0; m <= n; ++m) {
            float s = 0.f;
            #pragma unroll
            for (int d = 0; d < DIM; ++d) s += Qr[d] * s_K[kvBase + m * DIM + d];
            s *= rscale;
            sc[m] = s;
            mx = fmaxf(mx, s);
        }
        float sum = 0.f;
        for (int m = 0; m <= n; ++m) { const float e = __expf(sc[m] - mx); sc[m] = e; sum += e; }
        const float inv = 1.f / sum;
        float attn[DIM];
        #pragma unroll
        for (int d = 0; d < DIM; ++d) attn[d] = 0.f;
        for (int m = 0; m <= n; ++m) {
            const float w = sc[m] * inv;
            #pragma unroll
            for (int d = 0; d < DIM; ++d) attn[d] += w * s_V[kvBase + m * DIM + d];
        }
        // output projection + residual
        float X1[DIM];
        #pragma unroll
        for (int d = 0; d < DIM; ++d) {
            float o = s_bo[d];
            #pragma unroll
            for (int c = 0; c < DIM; ++c) o += attn[c] * s_Wo[d * DIM + c];
            X1[d] = Xr[d] + o;
        }
        // FFN + residual
        float X2[DIM];
        #pragma unroll
        for (int d = 0; d < DIM; ++d) X2[d] = X1[d] + s_b2[d];
        #pragma unroll
        for (int f = 0; f < FF; ++f) {
            float h = s_b1[f];
            #pragma unroll
            for (int c = 0; c < DIM; ++c) h += X1[c] * s_W1[f * DIM + c];
            h = fmaxf(h, 0.f);
            #pragma unroll
            for (int d = 0; d < DIM; ++d) X2[d] += h * s_W2[d * FF + f];
        }
        #pragma unroll
        for (int d = 0; d < DIM; ++d) s_X[tid * DIM + d] = X2[d];
    }
    __syncthreads();

    // ---- Phase 2: lm_head via v_wmma_f32_16x16x4_f32 ----
    // logits[256 x 128] = s_X[256 x 12] @ embed.T[12 x 128] + b_lm
    // K=12 -> three chained 16x16x4 WMMAs. 16 M-tiles x 8 N-tiles; 2 M-tiles per wave.
    const int lane   = tid & 31;
    const int wave   = tid >> 5;
    const int lane16 = lane & 15;
    const int laneHi = lane >> 4;

    for (int mt = wave * 2; mt < wave * 2 + 2; ++mt) {
        // A fragments (16x4 f32): lanes 0-15 hold K={k0,k0+1}, lanes 16-31 hold K={k0+2,k0+3}
        const int abase = (mt * 16 + lane16) * DIM + 2 * laneHi;
        v2f a0, a1, a2;
        a0.x = s_X[abase + 0]; a0.y = s_X[abase + 1];
        a1.x = s_X[abase + 4]; a1.y = s_X[abase + 5];
        a2.x = s_X[abase + 8]; a2.y = s_X[abase + 9];

        for (int nt = 0; nt < 8; ++nt) {
            // B fragments (4x16 f32): B(K,N) = embed[N][K]; same lane striping as A
            const int bbase = (nt * 16 + lane16) * DIM + 2 * laneHi;
            v2f b0v, b1v, b2v;
            b0v.x = s_embed[bbase + 0]; b0v.y = s_embed[bbase + 1];
            b1v.x = s_embed[bbase + 4]; b1v.y = s_embed[bbase + 5];
            b2v.x = s_embed[bbase + 8]; b2v.y = s_embed[bbase + 9];

            v8f acc = {};
            acc = __builtin_amdgcn_wmma_f32_16x16x4_f32(false, a0, false, b0v, (short)0, acc, false, false);
            acc = __builtin_amdgcn_wmma_f32_16x16x4_f32(false, a1, false, b1v, (short)0, acc, false, false);
            acc = __builtin_amdgcn_wmma_f32_16x16x4_f32(false, a2, false, b2v, (short)0, acc, false, false);

            const int   col  = nt * 16 + lane16;
            const float bias = s_blm[col];
            #pragma unroll
            for (int rI = 0; rI < 8; ++rI) {
                const int rowI = mt * 16 + rI + 8 * laneHi;   // D layout: VGPR r -> M=r (lanes 0-15), M=r+8 (lanes 16-31)
                out[(size_t)(tokBase + rowI) * VOC + col] = acc[rI] + bias;
            }
        }
    }
}

extern "C" void kernel_launch(void* const* d_in, const int* in_sizes, int n_in,
                              void* d_out, int out_size, void* d_ws, size_t ws_size,
                              hipStream_t stream) {
    const int*   x     = (const int*)  d_in[0];
    const float* embed = (const float*)d_in[1];
    const float* pos   = (const float*)d_in[2];
    const float* Wq    = (const float*)d_in[3];
    const float* bq    = (const float*)d_in[4];
    const float* Wk    = (const float*)d_in[5];
    const float* bk    = (const float*)d_in[6];
    const float* Wv    = (const float*)d_in[7];
    const float* bv    = (const float*)d_in[8];
    const float* Wo    = (const float*)d_in[9];
    const float* bo    = (const float*)d_in[10];
    const float* W1    = (const float*)d_in[11];
    const float* b1    = (const float*)d_in[12];
    const float* W2    = (const float*)d_in[13];
    const float* b2    = (const float*)d_in[14];
    const float* blm   = (const float*)d_in[15];
    float* out = (float*)d_out;

    const int batch  = in_sizes[0] / SEQ;           // 32768
    const int blocks = batch / ROWS_PER_BLOCK;      // 1024

    tiny_tf_kernel<<<blocks, TPB, 0, stream>>>(
        x, embed, pos, Wq, bq, Wk, bk, Wv, bv, Wo, bo, W1, b1, W2, b2, blm, out);
}